// OPTAttention_Mask_46136538694347
// MI455X (gfx1250) — compile-verified
//
#include <hip/hip_runtime.h>
#include <cstdint>
#include <cstddef>

#define S_LEN 2048
#define EMB   1024
#define NHEAD 16
#define HDIM  64
#define CB    408   // heavy + recent budget
#define HB    204   // heavy budget
#define RB    204   // recent budget
#define PEN   0.99f
#define NEGF  (-3.4028234663852886e38f)   // jnp.finfo(float32).min

typedef __attribute__((ext_vector_type(2))) float v2f;
typedef __attribute__((ext_vector_type(8))) float v8f;

// ---------------------------------------------------------------------------
// Async global->LDS support (gfx1250). Guarded: falls back to sync staging.
// Builtin signature (from hipcc diagnostic): param0 = int4* addrspace(1),
// param1 = int4* addrspace(3), then imm offset, imm cpol.
// ---------------------------------------------------------------------------
#if defined(__HIP_DEVICE_COMPILE__) && __has_builtin(__builtin_amdgcn_global_load_async_to_lds_b128)
#define HAVE_ASYNC 1
#else
#define HAVE_ASYNC 0
#endif

#if HAVE_ASYNC
typedef int v4i_t __attribute__((__vector_size__(4 * sizeof(int))));
typedef __attribute__((address_space(1))) v4i_t* as1_v4i;
typedef __attribute__((address_space(3))) v4i_t* as3_v4i;
__device__ __forceinline__ as1_v4i to_as1(const void* p) {
  return (as1_v4i)(unsigned long long)(uintptr_t)p;
}
__device__ __forceinline__ as3_v4i to_as3(const void* p) {
  // generic LDS address: low 32 bits are the DS byte offset (aperture map)
  return (as3_v4i)(unsigned int)(uintptr_t)p;
}
#if __has_builtin(__builtin_amdgcn_s_wait_asynccnt)
#define WAIT_ASYNC(n) __builtin_amdgcn_s_wait_asynccnt(n)
#else
#define WAIT_ASYNC(n) asm volatile("s_wait_asynccnt %0" ::"n"(n) : "memory")
#endif
#endif  // HAVE_ASYNC

// ---------------------------------------------------------------------------
// Generic NT GEMM with fp32 WMMA:  C[M,N] = A[M,K] @ Bt[N,K]^T  (+ epilogue)
// macro tile 128(M) x 64(N), K-step 32, 4 waves (128 thr), wave tile 64x32
// Double-buffered async LDS staging when available.
// ---------------------------------------------------------------------------
#define TM 128
#define TN 64
#define TK 32
#define LDSK (TK + 4)   // 36 floats: 16B-aligned rows, distinct bank residues

#if HAVE_ASYNC
#define NBUF 2
#else
#define NBUF 1
#endif

__global__ __launch_bounds__(128)
void gemm_nt_f32_wmma(const float* __restrict__ A, long lda, long strideA,
                      const float* __restrict__ Bt, long ldb, long strideB,
                      float* __restrict__ C, long ldc, long strideC,
                      int K,
                      const float* __restrict__ bias,
                      const float* __restrict__ amask, long mld) {
  __shared__ float sA[NBUF][TM][LDSK];
  __shared__ float sB[NBUF][TN][LDSK];

  const int tid   = threadIdx.x;
  const int lane  = tid & 31;
  const int wave  = tid >> 5;     // 0..3
  const int waveM = wave >> 1;    // 0..1 -> 64-row halves
  const int waveN = wave & 1;     // 0..1 -> 32-col halves
  const int lhalf = lane >> 4;    // 0/1
  const int lrow  = lane & 15;

  const int tileN = blockIdx.x * TN;
  const int tileM = blockIdx.y * TM;
  const size_t offA = (size_t)blockIdx.z * (size_t)strideA;
  const size_t offB = (size_t)blockIdx.z * (size_t)strideB;
  const size_t offC = (size_t)blockIdx.z * (size_t)strideC;

  v8f acc[4][2] = {};

  // 8 WMMA K-steps per staged tile; wave tile = 4x2 subtiles of 16x16
  auto compute_tile = [&](int cur) {
    #pragma unroll
    for (int kk = 0; kk < TK; kk += 4) {
      const int kb = kk + (lhalf << 1);
      v2f a[4], b[2];
      #pragma unroll
      for (int mi = 0; mi < 4; ++mi) {
        const int m = waveM * 64 + mi * 16 + lrow;
        a[mi].x = sA[cur][m][kb]; a[mi].y = sA[cur][m][kb + 1];
      }
      #pragma unroll
      for (int ni = 0; ni < 2; ++ni) {
        const int n = waveN * 32 + ni * 16 + lrow;
        b[ni].x = sB[cur][n][kb]; b[ni].y = sB[cur][n][kb + 1];
      }
      #pragma unroll
      for (int mi = 0; mi < 4; ++mi)
        #pragma unroll
        for (int ni = 0; ni < 2; ++ni)
          acc[mi][ni] = __builtin_amdgcn_wmma_f32_16x16x4_f32(
              false, a[mi], false, b[ni], (short)0, acc[mi][ni], false, false);
    }
  };

  const int nk = K / TK;

#if HAVE_ASYNC
  // per thread: 8 async b128 for A tile + 4 for B tile = 12 outstanding/tile
  auto issue_tile = [&](int kbase, int buf) {
    #pragma unroll
    for (int i = 0; i < 8; ++i) {
      const int id = tid + i * 128;
      const int r  = id >> 3;
      const int c4 = id & 7;
      const float* gp = A + offA + (size_t)(tileM + r) * (size_t)lda
                          + (size_t)(kbase + c4 * 4);
      __builtin_amdgcn_global_load_async_to_lds_b128(
          to_as1(gp), to_as3(&sA[buf][r][c4 * 4]), 0, 0);
    }
    #pragma unroll
    for (int i = 0; i < 4; ++i) {
      const int id = tid + i * 128;
      const int r  = id >> 3;
      const int c4 = id & 7;
      const float* gp = Bt + offB + (size_t)(tileN + r) * (size_t)ldb
                          + (size_t)(kbase + c4 * 4);
      __builtin_amdgcn_global_load_async_to_lds_b128(
          to_as1(gp), to_as3(&sB[buf][r][c4 * 4]), 0, 0);
    }
  };

  issue_tile(0, 0);
  for (int it = 0; it < nk; ++it) {
    const int cur = it & 1;
    if (it + 1 < nk) {
      issue_tile((it + 1) * TK, cur ^ 1);  // prefetch next tile into other buf
      WAIT_ASYNC(12);                      // in-order: previous tile landed
    } else {
      WAIT_ASYNC(0);
    }
    __syncthreads();
    compute_tile(cur);
    __syncthreads();
  }
#else
  for (int it = 0; it < nk; ++it) {
    const int k0 = it * TK;
    #pragma unroll
    for (int i = 0; i < 8; ++i) {
      int id = tid + i * 128;
      int r  = id >> 3;
      int c4 = id & 7;
      const float4 v = *(const float4*)(A + offA +
          (size_t)(tileM + r) * (size_t)lda + (size_t)(k0 + c4 * 4));
      sA[0][r][c4 * 4 + 0] = v.x; sA[0][r][c4 * 4 + 1] = v.y;
      sA[0][r][c4 * 4 + 2] = v.z; sA[0][r][c4 * 4 + 3] = v.w;
    }
    #pragma unroll
    for (int i = 0; i < 4; ++i) {
      int id = tid + i * 128;
      int r  = id >> 3;
      int c4 = id & 7;
      const float4 v = *(const float4*)(Bt + offB +
          (size_t)(tileN + r) * (size_t)ldb + (size_t)(k0 + c4 * 4));
      sB[0][r][c4 * 4 + 0] = v.x; sB[0][r][c4 * 4 + 1] = v.y;
      sB[0][r][c4 * 4 + 2] = v.z; sB[0][r][c4 * 4 + 3] = v.w;
    }
    if (k0 + TK < K) {
      __builtin_prefetch(A + offA + (size_t)(tileM + (tid >> 3)) * (size_t)lda
                           + (size_t)(k0 + TK), 0, 1);
      __builtin_prefetch(Bt + offB + (size_t)(tileN + ((tid >> 3) & 63)) * (size_t)ldb
                           + (size_t)(k0 + TK), 0, 1);
    }
    __syncthreads();
    compute_tile(0);
    __syncthreads();
  }
#endif

  // epilogue: C/D layout: VGPR j -> m = j + (lane>>4)*8, n = lane&15
  #pragma unroll
  for (int mi = 0; mi < 4; ++mi) {
    #pragma unroll
    for (int ni = 0; ni < 2; ++ni) {
      const int gm0 = tileM + waveM * 64 + mi * 16 + (lhalf << 3);
      const int gn  = tileN + waveN * 32 + ni * 16 + lrow;
      #pragma unroll
      for (int j = 0; j < 8; ++j) {
        const int gm = gm0 + j;
        float v = acc[mi][ni][j];
        if (bias)  v += bias[gn];
        if (amask) v = fmaxf(v + amask[(size_t)gm * (size_t)mld + gn], NEGF);
        C[offC + (size_t)gm * (size_t)ldc + gn] = v;
      }
    }
  }
}

// ---------------------------------------------------------------------------
// repack: head-major q (pre-scaled), k, and transposed vT[h][d][s]
// ---------------------------------------------------------------------------
__global__ __launch_bounds__(256)
void repack_qkv(const float* __restrict__ qlin, const float* __restrict__ klin,
                const float* __restrict__ vlin,
                float* __restrict__ q, float* __restrict__ k, float* __restrict__ vT) {
  const int idx = blockIdx.x * 256 + threadIdx.x;          // over H*S*D
  const int d = idx & (HDIM - 1);
  const int s = (idx >> 6) & (S_LEN - 1);
  const int h = idx >> 17;
  const size_t lin = (size_t)s * EMB + (size_t)h * HDIM + d;
  q[idx] = qlin[lin] * 0.125f;                              // D^-0.5 = 64^-0.5
  k[idx] = klin[lin];
  vT[((size_t)h * HDIM + d) * S_LEN + s] = vlin[lin];
}

// ---------------------------------------------------------------------------
// block reduction helpers (256 threads)
// ---------------------------------------------------------------------------
__device__ __forceinline__ float blk_max(float v, float* red, int tid) {
  red[tid] = v; __syncthreads();
  for (int o = 128; o > 0; o >>= 1) {
    if (tid < o) red[tid] = fmaxf(red[tid], red[tid + o]);
    __syncthreads();
  }
  float r = red[0]; __syncthreads(); return r;
}
__device__ __forceinline__ float blk_sum(float v, float* red, int tid) {
  red[tid] = v; __syncthreads();
  for (int o = 128; o > 0; o >>= 1) {
    if (tid < o) red[tid] += red[tid + o];
    __syncthreads();
  }
  float r = red[0]; __syncthreads(); return r;
}

// ---------------------------------------------------------------------------
// H2O heavy-hitter scan, fused with final masked softmax.
// One workgroup per head. attn[h][t][:] is overwritten with final probs.
// ---------------------------------------------------------------------------
__global__ __launch_bounds__(256)
void h2o_scan(float* __restrict__ attn) {
  const int tid = threadIdx.x;
  float* base = attn + (size_t)blockIdx.x * S_LEN * S_LEN;

  __shared__ float         s_acc[S_LEN];
  __shared__ unsigned char s_mask[S_LEN];
  __shared__ float         s_red[256];
  __shared__ unsigned int  s_hist[16];
  __shared__ unsigned int  s_scan[256];

  for (int s = tid; s < S_LEN; s += 256) {
    s_acc[s] = 0.f;
    s_mask[s] = (s < CB) ? 1 : 0;   // prev0 = cols < cb
  }
  __syncthreads();

  // ---- phase A: rows 0..CB-1: causal softmax in-place + penalty-weighted acc
  for (int t = 0; t < CB; ++t) {
    float* row = base + (size_t)t * S_LEN;
    float m = NEGF;
    for (int s = tid; s < S_LEN; s += 256) m = fmaxf(m, row[s]);
    const float rmax = blk_max(m, s_red, tid);
    float sum = 0.f;
    for (int s = tid; s < S_LEN; s += 256) sum += expf(row[s] - rmax);
    const float inv = 1.f / blk_sum(sum, s_red, tid);
    const float pf = powf(PEN, (float)(CB - 1 - t));
    for (int s = tid; s < S_LEN; s += 256) {
      const float p = expf(row[s] - rmax) * inv;
      row[s] = p;                 // final probs for this row
      s_acc[s] += pf * p;         // acc init einsum
    }
    __syncthreads();
  }

  // ---- phase B: sequential token loop
  for (int t = CB; t < S_LEN; ++t) {
    const int li = t - RB;        // local index

    // acc *= prev where col < li
    for (int s = tid; s < li; s += 256)
      if (!s_mask[s]) s_acc[s] = 0.f;
    __syncthreads();

    // exact top-HB of s_acc[0..li): 8-pass 4-bit radix select
    // (acc >= 0 so raw float bits are order-monotone)
    unsigned prefix = 0; int remaining = HB;
    for (int pass = 7; pass >= 0; --pass) {
      const int shift = pass * 4;
      if (tid < 16) s_hist[tid] = 0u;
      __syncthreads();
      for (int s = tid; s < li; s += 256) {
        const unsigned u = __float_as_uint(s_acc[s]);
        const bool match = (pass == 7) ||
                           ((u >> (shift + 4)) == (prefix >> (shift + 4)));
        if (match) atomicAdd(&s_hist[(u >> shift) & 15u], 1u);
      }
      __syncthreads();
      unsigned cum = 0; int bin = 0;
      for (int b = 15; b >= 0; --b) {
        cum += s_hist[b];
        if ((int)cum >= remaining) { bin = b; break; }
      }
      remaining -= (int)(cum - s_hist[bin]);
      prefix |= ((unsigned)bin) << shift;
      __syncthreads();
    }
    const unsigned T = prefix;
    const int eq_needed = remaining;   // #ties at T to keep (lowest index first)

    // tie-rank via contiguous chunks + block prefix scan (lowest-index wins)
    const int chunk = (li + 255) / 256;
    const int s0 = tid * chunk;
    int s1 = s0 + chunk; if (s1 > li) s1 = li;
    int cnt = 0;
    for (int s = s0; s < s1; ++s)
      if (__float_as_uint(s_acc[s]) == T) cnt++;
    s_scan[tid] = (unsigned)cnt; __syncthreads();
    for (int o = 1; o < 256; o <<= 1) {
      const unsigned x = s_scan[tid];
      const unsigned y = (tid >= o) ? s_scan[tid - o] : 0u;
      __syncthreads();
      s_scan[tid] = x + y;
      __syncthreads();
    }
    const int myoff = (int)s_scan[tid] - cnt;
    int r = 0;
    for (int s = s0; s < s1; ++s) {
      const unsigned u = __float_as_uint(s_acc[s]);
      const bool eq = (u == T);
      const bool heavy = (u > T) || (eq && (myoff + r) < eq_needed);
      if (eq) r++;
      s_mask[s] = heavy ? 1 : 0;
    }
    for (int s = li + tid; s < S_LEN; s += 256) s_mask[s] = 1;  // recent window
    __syncthreads();

    // masked softmax row (== final probs row) + acc update, in-place
    float* row = base + (size_t)t * S_LEN;
    float m = NEGF;
    for (int s = tid; s < S_LEN; s += 256) {
      const float v = s_mask[s] ? row[s] : NEGF;
      m = fmaxf(m, v);
    }
    const float rmax = blk_max(m, s_red, tid);
    float sum = 0.f;
    for (int s = tid; s < S_LEN; s += 256) {
      const float v = s_mask[s] ? row[s] : NEGF;
      sum += expf(v - rmax);
    }
    const float inv = 1.f / blk_sum(sum, s_red, tid);
    for (int s = tid; s < S_LEN; s += 256) {
      const float v = s_mask[s] ? row[s] : NEGF;
      const float p = expf(v - rmax) * inv;
      row[s] = p;
      s_acc[s] = s_acc[s] * PEN + p;
    }
    __syncthreads();
  }
}

// ---------------------------------------------------------------------------
extern "C" void kernel_launch(void* const* d_in, const int* in_sizes, int n_in,
                              void* d_out, int out_size, void* d_ws, size_t ws_size,
                              hipStream_t stream) {
  (void)in_sizes; (void)n_in; (void)out_size; (void)ws_size;
  const float* x  = (const float*)d_in[0];
  const float* am = (const float*)d_in[1];   // [1,1,S,S] causal mask values
  const float* Wq = (const float*)d_in[2];
  const float* bq = (const float*)d_in[3];
  const float* Wk = (const float*)d_in[4];
  const float* bk = (const float*)d_in[5];
  const float* Wv = (const float*)d_in[6];
  const float* bv = (const float*)d_in[7];
  const float* Wo = (const float*)d_in[8];
  const float* bo = (const float*)d_in[9];
  float* out = (float*)d_out;

  float* ws = (float*)d_ws;
  const size_t SEG = (size_t)S_LEN * EMB;   // 2M floats
  float* qlin = ws + 0 * SEG;
  float* klin = ws + 1 * SEG;
  float* vlin = ws + 2 * SEG;
  float* q    = ws + 3 * SEG;
  float* k    = ws + 4 * SEG;
  float* vT   = ws + 5 * SEG;
  float* ctx  = ws + 6 * SEG;
  float* attn = ws + 7 * SEG;               // H*S*S = 64M floats

  const dim3 blk(128);

  // QKV projections: [S,E] @ [E,E]^T + bias
  const dim3 gProj(EMB / TN, S_LEN / TM, 1);
  gemm_nt_f32_wmma<<<gProj, blk, 0, stream>>>(x, EMB, 0, Wq, EMB, 0,
                                              qlin, EMB, 0, EMB, bq, nullptr, 0);
  gemm_nt_f32_wmma<<<gProj, blk, 0, stream>>>(x, EMB, 0, Wk, EMB, 0,
                                              klin, EMB, 0, EMB, bk, nullptr, 0);
  gemm_nt_f32_wmma<<<gProj, blk, 0, stream>>>(x, EMB, 0, Wv, EMB, 0,
                                              vlin, EMB, 0, EMB, bv, nullptr, 0);

  repack_qkv<<<(NHEAD * S_LEN * HDIM) / 256, 256, 0, stream>>>(qlin, klin, vlin,
                                                               q, k, vT);

  // scores: per-head q @ k^T, + mask, clamp at NEG
  const dim3 gAttn(S_LEN / TN, S_LEN / TM, NHEAD);
  gemm_nt_f32_wmma<<<gAttn, blk, 0, stream>>>(
      q, HDIM, (long)S_LEN * HDIM, k, HDIM, (long)S_LEN * HDIM,
      attn, S_LEN, (long)S_LEN * S_LEN, HDIM, nullptr, am, S_LEN);

  // H2O scan (overwrites attn with final probs)
  h2o_scan<<<NHEAD, 256, 0, stream>>>(attn);

  // probs @ V: per-head [S,S] @ vT[D,S]^T -> ctx[s][h*64+d]
  const dim3 gPV(HDIM / TN, S_LEN / TM, NHEAD);
  gemm_nt_f32_wmma<<<gPV, blk, 0, stream>>>(
      attn, S_LEN, (long)S_LEN * S_LEN, vT, S_LEN, (long)HDIM * S_LEN,
      ctx, EMB, HDIM, S_LEN, nullptr, nullptr, 0);

  // output projection
  const dim3 gOut(EMB / TN, S_LEN / TM, 1);
  gemm_nt_f32_wmma<<<gOut, blk, 0, stream>>>(ctx, EMB, 0, Wo, EMB, 0,
                                             out, EMB, 0, EMB, bo, nullptr, 0);
}